// GNNTower_75222057222468
// MI455X (gfx1250) — compile-verified
//
#include <hip/hip_runtime.h>

#define NN 50000
#define NE 800000
#define DIM 128
#define NG 64
#define PS2 144   // float2 stride per LDS row-pair: dword stride 288 = 32 mod 64 -> conflict-free

typedef __attribute__((ext_vector_type(2))) float v2f;
typedef __attribute__((ext_vector_type(8))) float v8f;

// ---------------- small utility kernels ----------------

__global__ void zero_i(int* __restrict__ p, int n) {
    int i = blockIdx.x * blockDim.x + threadIdx.x;
    if (i < n) p[i] = 0;
}

__global__ void count_i(const int* __restrict__ idx, int* __restrict__ cnt, int n) {
    int i = blockIdx.x * blockDim.x + threadIdx.x;
    if (i < n) atomicAdd(&cnt[idx[i]], 1);
}

__global__ void inv_from_count(const int* __restrict__ cnt, float* __restrict__ inv, int n) {
    int i = blockIdx.x * blockDim.x + threadIdx.x;
    if (i < n) inv[i] = 1.0f / (float)max(cnt[i], 1);
}

// one-block scan of per-node degree -> CSR row_start + scatter cursor
__global__ __launch_bounds__(1024)
void scan_deg(const int* __restrict__ deg, int* __restrict__ row_start,
              int* __restrict__ cursor, int n) {
    __shared__ int part[1024];
    const int tid = threadIdx.x;
    const int chunk = (n + 1023) >> 10;
    int lo = min(tid * chunk, n);
    int hi = min(lo + chunk, n);
    int s = 0;
    for (int i = lo; i < hi; ++i) s += deg[i];
    part[tid] = s;
    __syncthreads();
    for (int off = 1; off < 1024; off <<= 1) {   // inclusive Hillis-Steele
        int v = part[tid];
        int add = (tid >= off) ? part[tid - off] : 0;
        __syncthreads();
        part[tid] = v + add;
        __syncthreads();
    }
    int base = (tid == 0) ? 0 : part[tid - 1];
    for (int i = lo; i < hi; ++i) {
        row_start[i] = base;
        cursor[i] = base;
        base += deg[i];
    }
    if (tid == 1023) row_start[n] = part[1023];
}

// tiny serial scan over 64 graph counts (batch_vec is sorted -> contiguous ranges)
__global__ void graph_starts(const int* __restrict__ gcnt, int* __restrict__ gstart) {
    if (threadIdx.x == 0 && blockIdx.x == 0) {
        int acc = 0;
        for (int g = 0; g < NG; ++g) { gstart[g] = acc; acc += gcnt[g]; }
        gstart[NG] = acc;
    }
}

// bucket edges by dst (once per launch; cheap int atomics)
__global__ void edge_bucket(const int* __restrict__ src, const int* __restrict__ dst,
                            int* __restrict__ cursor, int* __restrict__ ssrc, int nE) {
    int e = blockIdx.x * blockDim.x + threadIdx.x;
    if (e < nE) {
        int pos = atomicAdd(&cursor[dst[e]], 1);
        ssrc[pos] = src[e];
    }
}

// ---------------- CSR gather aggregation (bandwidth-bound phase) ----------------
// One wave per dst node; lane owns a float4 channel chunk -> each edge is one
// coalesced 512B row read. No float atomics, no agg zeroing; deg-normalize inline.

__global__ void aggregate(const float* __restrict__ x, const int* __restrict__ row_start,
                          const int* __restrict__ ssrc, const float* __restrict__ inv_deg,
                          float* __restrict__ agg, int n) {
    int w = (blockIdx.x * blockDim.x + threadIdx.x) >> 5;
    if (w >= n) return;
    const int c = (threadIdx.x & 31) << 2;
    const int beg = row_start[w];
    const int end = row_start[w + 1];
    float4 acc = {0.f, 0.f, 0.f, 0.f};
    for (int j = beg; j < end; ++j) {
        int s = ssrc[j];
        const float4 v = *reinterpret_cast<const float4*>(x + (size_t)s * DIM + c);
        acc.x += v.x; acc.y += v.y; acc.z += v.z; acc.w += v.w;
    }
    const float idg = inv_deg[w];
    float4 o = {acc.x * idg, acc.y * idg, acc.z * idg, acc.w * idg};
    *reinterpret_cast<float4*>(agg + (size_t)w * DIM + c) = o;
}

// ---------------- fused layer GEMM with fp32 WMMA ----------------
// Y = relu(X @ Ws + AggN @ Wn + bs + bn), AggN pre-normalized.
// Wave -> 16 rows x 128 cols; K stepped by 4 via v_wmma_f32_16x16x4_f32.
// A fragments preloaded into registers (32 b64 loads in flight, no per-chunk stall).
// W staged in LDS pair-interleaved: B fragment = single conflict-free ds_load_b64.

__global__ __launch_bounds__(128)
void gnn_layer(const float* __restrict__ X, const float* __restrict__ Agg,
               const float* __restrict__ Ws, const float* __restrict__ Wn,
               const float* __restrict__ bs, const float* __restrict__ bn,
               float* __restrict__ Y, int nRows) {
    extern __shared__ float smem[];                 // 64*PS2 float2 + 128 floats
    float2* w2 = reinterpret_cast<float2*>(smem);
    float* b_lds = smem + 64 * PS2 * 2;

    const int tid     = threadIdx.x;
    const int lane    = tid & 31;
    const int wave    = tid >> 5;
    const int halfsel = lane >> 4;
    const int l15     = lane & 15;

    if (tid < DIM) b_lds[tid] = bs[tid] + bn[tid];

    const int row_base = blockIdx.x * 64 + wave * 16;
    int arow = row_base + l15;
    if (arow >= nRows) arow = nRows - 1;            // clamp loads; stores guarded

    // stage Ws pair-interleaved: w2[p*PS2 + c] = {Ws[2p][c], Ws[2p+1][c]}
    for (int i = tid; i < 64 * DIM; i += 128) {
        int p = i >> 7, c = i & (DIM - 1);
        float2 t;
        t.x = Ws[(size_t)(2 * p) * DIM + c];
        t.y = Ws[(size_t)(2 * p + 1) * DIM + c];
        w2[p * PS2 + c] = t;
    }
    __syncthreads();

    v8f acc[8];
#pragma unroll
    for (int n = 0; n < 8; ++n) {
        float bb = b_lds[n * 16 + l15];
        v8f t = {bb, bb, bb, bb, bb, bb, bb, bb};
        acc[n] = t;
    }

    // ---- phase 1: X @ Ws ----
    {
        const float2* Arow2 = reinterpret_cast<const float2*>(X + (size_t)arow * DIM);
        v2f areg[32];
#pragma unroll
        for (int kc = 0; kc < 32; ++kc) {
            float2 t = Arow2[2 * kc + halfsel];
            v2f a; a.x = t.x; a.y = t.y;
            areg[kc] = a;
        }
#pragma unroll
        for (int kc = 0; kc < 32; ++kc) {
            const float2* wrow = w2 + (2 * kc + halfsel) * PS2 + l15;
#pragma unroll
            for (int n = 0; n < 8; ++n) {
                float2 bt = wrow[n * 16];
                v2f b; b.x = bt.x; b.y = bt.y;
                acc[n] = __builtin_amdgcn_wmma_f32_16x16x4_f32(
                    false, areg[kc], false, b, (short)0, acc[n], false, false);
            }
        }
    }

    // ---- restage Wn ----
    __syncthreads();
    for (int i = tid; i < 64 * DIM; i += 128) {
        int p = i >> 7, c = i & (DIM - 1);
        float2 t;
        t.x = Wn[(size_t)(2 * p) * DIM + c];
        t.y = Wn[(size_t)(2 * p + 1) * DIM + c];
        w2[p * PS2 + c] = t;
    }
    __syncthreads();

    // ---- phase 2: AggN @ Wn ----
    {
        const float2* Grow2 = reinterpret_cast<const float2*>(Agg + (size_t)arow * DIM);
        v2f areg[32];
#pragma unroll
        for (int kc = 0; kc < 32; ++kc) {
            float2 t = Grow2[2 * kc + halfsel];
            v2f a; a.x = t.x; a.y = t.y;
            areg[kc] = a;
        }
#pragma unroll
        for (int kc = 0; kc < 32; ++kc) {
            const float2* wrow = w2 + (2 * kc + halfsel) * PS2 + l15;
#pragma unroll
            for (int n = 0; n < 8; ++n) {
                float2 bt = wrow[n * 16];
                v2f b; b.x = bt.x; b.y = bt.y;
                acc[n] = __builtin_amdgcn_wmma_f32_16x16x4_f32(
                    false, areg[kc], false, b, (short)0, acc[n], false, false);
            }
        }
    }

    // ---- relu + store ----
#pragma unroll
    for (int n = 0; n < 8; ++n) {
#pragma unroll
        for (int j = 0; j < 8; ++j) {
            int row = row_base + j + 8 * halfsel;
            if (row < nRows) {
                Y[(size_t)row * DIM + n * 16 + l15] = fmaxf(acc[n][j], 0.0f);
            }
        }
    }
}

// ---------------- gather-based mean pooling (batch_vec sorted) ----------------

__global__ __launch_bounds__(128)
void pool_mean(const float* __restrict__ x, const int* __restrict__ gstart,
               float* __restrict__ out) {
    const int g = blockIdx.x;
    const int c = threadIdx.x;                       // 128 channels
    const int beg = gstart[g];
    const int end = gstart[g + 1];
    float acc = 0.0f;
    for (int r = beg; r < end; ++r) acc += x[(size_t)r * DIM + c];
    out[(size_t)g * DIM + c] = acc / fmaxf((float)(end - beg), 1.0f);
}

// ---------------- launch ----------------

extern "C" void kernel_launch(void* const* d_in, const int* in_sizes, int n_in,
                              void* d_out, int out_size, void* d_ws, size_t ws_size,
                              hipStream_t stream) {
    const float* x       = (const float*)d_in[0];   // [NN, DIM]
    const float* w_self  = (const float*)d_in[1];   // [3, DIM, DIM]
    const float* b_self  = (const float*)d_in[2];   // [3, DIM]
    const float* w_neigh = (const float*)d_in[3];   // [3, DIM, DIM]
    const float* b_neigh = (const float*)d_in[4];   // [3, DIM]
    const int*   eidx    = (const int*)d_in[5];     // [2, NE]
    const int*   batch   = (const int*)d_in[6];     // [NN]
    const int* src = eidx;
    const int* dst = eidx + NE;
    float* out = (float*)d_out;                     // [NG, DIM]

    // workspace layout (4-byte units)
    int*   wsi = (int*)d_ws;
    float* wsf = (float*)d_ws;
    int*   deg_i    = wsi + 0;                       // 50176
    float* inv_deg  = wsf + 50176;                   // 50176
    int*   row_st   = wsi + 100352;                  // 50176 (>= NN+1)
    int*   cursor   = wsi + 150528;                  // 50176
    int*   gcnt     = wsi + 200704;                  // 128
    int*   gstart   = wsi + 200832;                  // 128 (>= NG+1)
    int*   ssrc     = wsi + 200960;                  // NE
    float* buf0     = wsf + 1000960;                 // NN*DIM
    float* buf1     = buf0 + (size_t)NN * DIM;
    float* agg      = buf1 + (size_t)NN * DIM;

    const int ZB = 256;
    const int gemm_blocks = (NN + 63) / 64;
    const size_t gemm_lds = (size_t)(64 * PS2 * 2 + DIM) * sizeof(float);

    // --- CSR build (once per launch; counters re-zeroed every call) ---
    zero_i<<<(NN + ZB - 1) / ZB, ZB, 0, stream>>>(deg_i, NN);
    zero_i<<<1, ZB, 0, stream>>>(gcnt, NG);
    count_i<<<(NE + ZB - 1) / ZB, ZB, 0, stream>>>(dst, deg_i, NE);
    count_i<<<(NN + ZB - 1) / ZB, ZB, 0, stream>>>(batch, gcnt, NN);
    inv_from_count<<<(NN + ZB - 1) / ZB, ZB, 0, stream>>>(deg_i, inv_deg, NN);
    scan_deg<<<1, 1024, 0, stream>>>(deg_i, row_st, cursor, NN);
    graph_starts<<<1, 64, 0, stream>>>(gcnt, gstart);
    edge_bucket<<<(NE + ZB - 1) / ZB, ZB, 0, stream>>>(src, dst, cursor, ssrc, NE);

    // --- 3 layers: gather-aggregate + WMMA GEMM ---
    const float* xin = x;
    float* bufs[3] = {buf0, buf1, buf0};
    for (int l = 0; l < 3; ++l) {
        aggregate<<<(NN * 32 + ZB - 1) / ZB, ZB, 0, stream>>>(
            xin, row_st, ssrc, inv_deg, agg, NN);
        gnn_layer<<<gemm_blocks, 128, gemm_lds, stream>>>(
            xin, agg,
            w_self + (size_t)l * DIM * DIM, w_neigh + (size_t)l * DIM * DIM,
            b_self + (size_t)l * DIM, b_neigh + (size_t)l * DIM,
            bufs[l], NN);
        xin = bufs[l];
    }

    // --- mean pooling over sorted batch ranges (no atomics, fully overwrites out) ---
    pool_mean<<<NG, 128, 0, stream>>>(xin, gstart, out);
}